// ITCEDistribution_20658792693976
// MI455X (gfx1250) — compile-verified
//
#include <hip/hip_runtime.h>

// ---------------------------------------------------------------------------
// ITCE distribution: log_prob + entropy of N(mean, Sigma) with
//   Sigma_b = C diag(h2_b @ corr_var) C^T + diag(h2_b @ ind_var)
// Restructured:  Sigma_b[a,m] = sum_l h2_b[l] * M[l, a*64+m],
//   M[l,am] = sum_j corr_var[l,j] C[a,j] C[m,j]   (one-time, 8.5 MB, L2-hot)
// Main cost: one fp32 WMMA GEMM [32768x512] x [512x4160]  (~140 GFLOP fp32).
// M is stored k-chunk swizzled: Msw[c][n][k%16] (c = k/16) so that
//  - each 16x64 B panel chunk is one contiguous 4 KB block  -> async LDS copy
//  - B fragments are contiguous b64 LDS reads (no v_mov marshaling)
// ---------------------------------------------------------------------------

#define LATENT   512
#define ADIM     64
#define BATCH_N  32768
#define NCOLS    (ADIM * ADIM + ADIM)   // 4160 = sigma(4096) | v_ind(64)

#define LOG2PI_F        1.8378770664093453f
#define LOG_DIM_CORR_F  3.1191623125197542f   // 0.5*ln(512)
#define LOG_MIN_STD_F  (-6.907755278982137f)  // ln(0.001)
#define LOG_MAX_STD_F   4.605170185988092f    // ln(100)

typedef float v2f __attribute__((ext_vector_type(2)));
typedef float v8f __attribute__((ext_vector_type(8)));

// CDNA5 wave-synchronous LDS fence: wait for outstanding DS ops, pin ordering.
__device__ __forceinline__ void wave_lds_fence() {
  asm volatile("s_wait_dscnt 0x0" ::: "memory");
  __builtin_amdgcn_wave_barrier();
}

// Async global->LDS copy (ASYNCcnt-tracked), GVS addressing:
//   each lane copies 16 bytes from global address base + goff to the LDS address in lds_off.
__device__ __forceinline__ void async_load_b128(unsigned lds_off, int goff,
                                                const void* base) {
  asm volatile("global_load_async_to_lds_b128 %0, %1, %2"
               :: "v"(lds_off), "v"(goff), "s"(base) : "memory");
}
__device__ __forceinline__ void wait_async0() {
  asm volatile("s_wait_asynccnt 0x0" ::: "memory");
}

__device__ __forceinline__ float clamped_var(float ls) {
  ls = fminf(fmaxf(ls, LOG_MIN_STD_F), LOG_MAX_STD_F) - LOG_DIM_CORR_F;
  return __expf(2.0f * ls);
}

// ---------------------------------------------------------------------------
// Kernel 1: Msw[(c*NCOLS + n)*16 + kk] = M[l = c*16+kk, n]
//   n < 4096:  M[l, a*64+m] = sum_j corr_var[l,j]*C[a,j]*C[m,j]
//   n >= 4096: ind_var[l, n-4096]
// One block per l (512 blocks). ~1 G MAC one-time.
// ---------------------------------------------------------------------------
__global__ __launch_bounds__(256)
void k1_build_M(const float* __restrict__ log_std,   // [512, 576]
                const float* __restrict__ cov_w,     // [64, 512]
                float* __restrict__ Msw)             // [32][4160][16] floats
{
  __shared__ float cv[LATENT];
  const int l = blockIdx.x;
  const int tid = threadIdx.x;
  for (int j = tid; j < LATENT; j += 256)
    cv[j] = clamped_var(log_std[(size_t)l * (LATENT + ADIM) + j]);
  __syncthreads();

  const size_t swz = ((size_t)(l >> 4) * NCOLS) * 16 + (l & 15);
  for (int n = tid; n < NCOLS; n += 256) {
    float r;
    if (n < ADIM * ADIM) {
      const float* Ca = cov_w + (size_t)(n >> 6) * LATENT;
      const float* Cm = cov_w + (size_t)(n & 63) * LATENT;
      float s = 0.0f;
      for (int j = 0; j < LATENT; ++j) s = fmaf(cv[j] * Ca[j], Cm[j], s);
      r = s;
    } else {
      r = clamped_var(log_std[(size_t)l * (LATENT + ADIM) + LATENT + (n - ADIM * ADIM)]);
    }
    Msw[swz + (size_t)n * 16] = r;
  }
}

// ---------------------------------------------------------------------------
// Kernel 2: fp32 WMMA GEMM: Sig[b,n] = sum_l latent[b,l]^2 * M[l,n]
// Macro-tile 256x64, 8 waves as 4(row)x2(col), each wave 4x2 WMMA tiles.
// B panel chunks arrive via global_load_async_to_lds_b128 (ASYNCcnt).
// ---------------------------------------------------------------------------
__global__ __launch_bounds__(256, 2)
void k2_sigma_gemm(const float* __restrict__ latent,  // [B, 512]
                   const float* __restrict__ Msw,     // swizzled [32][4160][16]
                   float* __restrict__ Sig)           // [B, 4160]
{
  __shared__ __align__(16) float As[256][16];   // h2 tile, k contiguous (16 KB)
  __shared__ __align__(16) float Bt[64][16];    // M tile, k contiguous (4 KB)

  const int tid  = threadIdx.x;
  const int lane = tid & 31;
  const int wave = tid >> 5;
  const int wr   = (wave & 3) * 64;   // wave row offset in macro-tile
  const int wc   = (wave >> 2) * 32;  // wave col offset in macro-tile
  const int row0 = blockIdx.y * 256;
  const int col0 = blockIdx.x * 64;
  const int lhi  = lane >> 4;         // half-wave: K-pair select
  const int llo  = lane & 15;

  const v8f vzero = {};
  v8f acc[4][2];
#pragma unroll
  for (int r = 0; r < 4; ++r)
#pragma unroll
    for (int c = 0; c < 2; ++c) acc[r][c] = vzero;

  const unsigned bt_lds = (unsigned)(size_t)(&Bt[0][0]) + tid * 16u; // LDS byte offset

  for (int k0 = 0; k0 < LATENT; k0 += 16) {
    // --- async stage B: contiguous 4 KB chunk of swizzled M ---------------
    const int goff = (((k0 >> 4) * NCOLS + col0) * 16 + tid * 4) * 4;  // bytes
    async_load_b128(bt_lds, goff, Msw);

    // --- stage A (squared on the fly): 256x16 floats, 4x float4/thread ----
#pragma unroll
    for (int q = 0; q < 4; ++q) {
      const int idx = tid + q * 256;        // float4 index in 256x4 grid
      const int ar  = idx >> 2;
      const int akq = (idx & 3) * 4;
      float4 a = *(const float4*)(latent + (size_t)(row0 + ar) * LATENT + k0 + akq);
      float4 h2 = make_float4(a.x * a.x, a.y * a.y, a.z * a.z, a.w * a.w);
      *(float4*)&As[ar][akq] = h2;
    }
    if (k0 + 16 < LATENT)
      __builtin_prefetch(latent + (size_t)(row0 + (tid >> 2) + ((tid & 2) << 6)) * LATENT + k0 + 16, 0, 1);

    wait_async0();
    __syncthreads();

    // --- 32 WMMAs per k-chunk --------------------------------------------
#pragma unroll
    for (int ks = 0; ks < 16; ks += 4) {
      const int kk = ks + 2 * lhi;  // K pair owned by this half-wave
      v2f af[4], bf[2];
#pragma unroll
      for (int r = 0; r < 4; ++r) af[r] = *(const v2f*)&As[wr + r * 16 + llo][kk];
#pragma unroll
      for (int c = 0; c < 2; ++c) bf[c] = *(const v2f*)&Bt[wc + c * 16 + llo][kk];
#pragma unroll
      for (int r = 0; r < 4; ++r)
#pragma unroll
        for (int c = 0; c < 2; ++c)
          acc[r][c] = __builtin_amdgcn_wmma_f32_16x16x4_f32(
              false, af[r], false, bf[c], (short)0, acc[r][c], false, false);
    }
    __syncthreads();
  }

  // C/D layout: VGPR v, lanes 0-15 -> M=v, lanes 16-31 -> M=v+8; N=llo.
#pragma unroll
  for (int r = 0; r < 4; ++r) {
#pragma unroll
    for (int v = 0; v < 8; ++v) {
      const int rr = wr + r * 16 + v + 8 * lhi;
      const size_t o = (size_t)(row0 + rr) * NCOLS + col0 + wc + llo;
      Sig[o]      = acc[r][0][v];
      Sig[o + 16] = acc[r][1][v];
    }
  }
}

// ---------------------------------------------------------------------------
// Kernel 3: per batch row -- add v_ind to diag, mean/diff, 64x64 Cholesky in
// LDS (stride 65 vs 64 banks), forward solve, log_prob + entropy.
// One wave32 per row, 8 rows per workgroup (LDS ~133 KB of 320 KB).
// ---------------------------------------------------------------------------
#define ROWS3 8
#define SW    65

__global__ __launch_bounds__(256, 1)
void k3_chol_logprob(const float* __restrict__ Sig,     // [B, 4160]
                     const float* __restrict__ latent,  // [B, 512]
                     const float* __restrict__ actions, // [B, 64]
                     const float* __restrict__ mean_w,  // [64, 512]
                     const float* __restrict__ mean_b,  // [64]
                     float* __restrict__ out)           // [2, B]
{
  __shared__ float Sh[ROWS3][ADIM][SW];
  __shared__ float Dh[ROWS3][ADIM];

  const int lane = threadIdx.x & 31;
  const int wave = threadIdx.x >> 5;
  const int b    = blockIdx.x * ROWS3 + wave;

  float* Sm = &Sh[wave][0][0];
  float* d  = &Dh[wave][0];
  const float* sig = Sig + (size_t)b * NCOLS;
  const float* x   = latent + (size_t)b * LATENT;

  // Load sigma, add v_ind to diagonal.
  for (int t = lane; t < ADIM * ADIM; t += 32)
    Sm[(t >> 6) * SW + (t & 63)] = sig[t];
  wave_lds_fence();
  for (int i = lane; i < ADIM; i += 32)
    Sm[i * SW + i] += sig[ADIM * ADIM + i];

  // diff = actions - (mean_w @ x + mean_b); coalesced + shuffle reduce.
  for (int a = 0; a < ADIM; ++a) {
    const float* w = mean_w + (size_t)a * LATENT;
    float s = 0.0f;
    for (int j = lane; j < LATENT; j += 32) s = fmaf(w[j], x[j], s);
#pragma unroll
    for (int off = 16; off > 0; off >>= 1) s += __shfl_down(s, off, 32);
    if (lane == 0) d[a] = actions[(size_t)b * ADIM + a] - (s + mean_b[a]);
  }
  wave_lds_fence();

  // In-place lower Cholesky.
  float logdet = 0.0f;
  for (int k = 0; k < ADIM; ++k) {
    const float dkk = sqrtf(Sm[k * SW + k]);  // same value in all lanes
    logdet += __logf(dkk);
    const float inv = 1.0f / dkk;
    for (int i = k + 1 + lane; i < ADIM; i += 32) Sm[i * SW + k] *= inv;
    if (lane == 0) Sm[k * SW + k] = dkk;
    wave_lds_fence();
    for (int i = k + 1; i < ADIM; ++i) {
      const float lik = Sm[i * SW + k];
      for (int j = k + 1 + lane; j <= i; j += 32)
        Sm[i * SW + j] -= lik * Sm[j * SW + k];
    }
    wave_lds_fence();
  }

  // Forward solve L z = diff; maha = |z|^2.
  float maha = 0.0f;
  for (int i = 0; i < ADIM; ++i) {
    const float zi = d[i] / Sm[i * SW + i];
    maha += zi * zi;
    for (int j = i + 1 + lane; j < ADIM; j += 32)
      d[j] -= Sm[j * SW + i] * zi;
    wave_lds_fence();
  }

  logdet *= 2.0f;
  if (lane == 0) {
    out[b]           = -0.5f * (maha + logdet + (float)ADIM * LOG2PI_F);
    out[BATCH_N + b] =  0.5f * ((float)ADIM * (1.0f + LOG2PI_F) + logdet);
  }
}

// ---------------------------------------------------------------------------
extern "C" void kernel_launch(void* const* d_in, const int* in_sizes, int n_in,
                              void* d_out, int out_size, void* d_ws, size_t ws_size,
                              hipStream_t stream) {
  (void)in_sizes; (void)n_in; (void)out_size; (void)ws_size;
  const float* latent  = (const float*)d_in[0];
  const float* actions = (const float*)d_in[1];
  const float* mean_w  = (const float*)d_in[2];
  const float* mean_b  = (const float*)d_in[3];
  const float* cov_w   = (const float*)d_in[4];
  const float* log_std = (const float*)d_in[5];
  float* out = (float*)d_out;

  // Workspace layout: Msw [512*4160] (8.5 MB) | Sig [32768*4160] (545 MB).
  float* Msw = (float*)d_ws;
  float* Sig = Msw + (size_t)LATENT * NCOLS;

  k1_build_M<<<dim3(LATENT), dim3(256), 0, stream>>>(log_std, cov_w, Msw);
  k2_sigma_gemm<<<dim3(NCOLS / 64, BATCH_N / 256), dim3(256), 0, stream>>>(latent, Msw, Sig);
  k3_chol_logprob<<<dim3(BATCH_N / ROWS3), dim3(256), 0, stream>>>(Sig, latent, actions,
                                                                   mean_w, mean_b, out);
}